// NaiveWaveletKANLayer_1743756722947
// MI455X (gfx1250) — compile-verified
//
#include <hip/hip_runtime.h>

typedef __attribute__((ext_vector_type(2))) float v2f;
typedef __attribute__((ext_vector_type(4))) float v4f;
typedef __attribute__((ext_vector_type(8))) float v8f;

#if __has_builtin(__builtin_amdgcn_exp2f)
#define FAST_EXP2(x) __builtin_amdgcn_exp2f(x)   // v_exp_f32, no lib wrapper
#else
#define FAST_EXP2(x) exp2f(x)                    // device overload fallback
#endif

namespace {
constexpr int kB      = 16384;
constexpr int kInDim  = 256;
constexpr int kOutDim = 64;
constexpr int kGrid   = 16;

constexpr int kRowTiles   = 2;                         // 2 x 16 rows per wave = 32
constexpr int kColTiles   = 4;                         // 4 x 16 cols = all 64 outputs
constexpr int kWavesPerWg = 8;                         // 256 threads / wave32
constexpr int kRowsPerWg  = kWavesPerWg * kRowTiles * 16;   // 256 rows per block
constexpr int kSlabs      = 8;                         // 32 K-values per i = 8 slabs of K=4
// packed weights: pw[(((i*8 + j)*4 + t)*32 + lane)*2 + v]
constexpr int kPackElems  = kInDim * kSlabs * kColTiles * 64;   // 524288 floats (2 MB)
}

// ---------------------------------------------------------------------------
// Repack waveletcoeffs[part][o][i][g] (fp32) into per-lane WMMA B-fragment
// order for V_WMMA_F32_16X16X4_F32.  B is 4x16 per fragment; VGPR v holds
// K_local = v + 2*(lane>=16), N = t*16 + (lane&15).  Global K = i*32 + g*2 + part.
// ---------------------------------------------------------------------------
__global__ void pack_weights_kernel(const float* __restrict__ wc,
                                    float* __restrict__ pw) {
  int idx = blockIdx.x * blockDim.x + threadIdx.x;
  if (idx >= kPackElems) return;
  int v    = idx & 1;
  int lane = (idx >> 1) & 31;
  int t    = (idx >> 6) & 3;
  int j    = (idx >> 8) & 7;
  int i    = idx >> 11;
  int half = lane >> 4;
  int kl   = v + 2 * half;            // K_local in [0,4)
  int g    = 2 * j + (kl >> 1);       // grid index
  int part = kl & 1;                  // 0 = real coeffs, 1 = imag coeffs
  int n    = t * 16 + (lane & 15);    // output column
  pw[idx] = wc[((size_t)(part * kOutDim + n) * kInDim + i) * kGrid + g];
}

// ---------------------------------------------------------------------------
// Main kernel: per wave, 32 rows x 64 cols of the output.  For each input dim
// i and K-slab j, every lane evaluates ONE Morlet wavelet (exp2+sin+cos) which
// is exactly its fp32 A-fragment (v0=real, v1=imag, g = 2j + lane/16), then
// feeds 8 accumulators via v_wmma_f32_16x16x4_f32.
// ---------------------------------------------------------------------------
__global__ __launch_bounds__(kWavesPerWg * 32) void wavelet_kan_wmma_kernel(
    const float* __restrict__ x, const float* __restrict__ pw,
    const float* __restrict__ bias, float* __restrict__ out) {
  const int tid  = threadIdx.x;
  const int lane = tid & 31;
  const int wave = tid >> 5;
  const int l16  = lane & 15;
  const int half = lane >> 4;
  const int rowbase = blockIdx.x * kRowsPerWg + wave * (kRowTiles * 16);

  constexpr float kPi      = 3.14159265358979323846f;
  constexpr float kInvG15  = 1.0f / 15.0f;        // translations = linspace(0,1,16)
  constexpr float kNegHL2e = -0.72134752044448f;  // -0.5 * log2(e)

  // Hoisted per-lane wavelet constants for each of the 8 K-slabs:
  // this lane's grid index in slab j is g = 2j + half.
  float sgp[kSlabs], tgp[kSlabs];
#pragma unroll
  for (int j = 0; j < kSlabs; ++j) {
    float gf = (float)(2 * j + half);
    sgp[j] = gf + 1.0f;       // scales = linspace(1,16,16)
    tgp[j] = gf * kInvG15;    // translations
  }

  v8f acc[kRowTiles][kColTiles];
#pragma unroll
  for (int r = 0; r < kRowTiles; ++r)
#pragma unroll
    for (int t = 0; t < kColTiles; ++t)
      acc[r][t] = {};

  float bv[kColTiles];
#pragma unroll
  for (int t = 0; t < kColTiles; ++t) bv[t] = bias[t * 16 + l16];

  for (int i4 = 0; i4 < kInDim / 4; ++i4) {
    // x values for the 4 upcoming input dims (lane pairs L and L+16 share rows)
    v4f xv[kRowTiles];
#pragma unroll
    for (int r = 0; r < kRowTiles; ++r)
      xv[r] = *(const v4f*)(x + (size_t)(rowbase + r * 16 + l16) * kInDim + i4 * 4);

    // stream next weight block toward the caches (global_prefetch_b8)
    if (i4 + 1 < kInDim / 4)
      __builtin_prefetch(pw + (size_t)(i4 + 1) * 4 * kSlabs * kColTiles * 64 +
                             lane * 16, 0, 0);

#pragma unroll
    for (int ii = 0; ii < 4; ++ii) {
      const int i = i4 * 4 + ii;
#pragma unroll
      for (int j = 0; j < kSlabs; ++j) {
        // B fragments: coalesced 8B/lane from the packed weight stream
        const float* fb = pw + ((size_t)(i * kSlabs + j) * kColTiles) * 64;
        v2f bf[kColTiles];
#pragma unroll
        for (int t = 0; t < kColTiles; ++t)
          bf[t] = *(const v2f*)(fb + t * 64 + lane * 2);

        // A fragments: one wavelet evaluation per lane per row-tile.
        // env = exp(-u^2/2) computed as exp2(u^2 * -0.5*log2(e)): 1 fewer mul.
        v2f af[kRowTiles];
#pragma unroll
        for (int r = 0; r < kRowTiles; ++r) {
          float u  = (xv[r][ii] - tgp[j]) * sgp[j];
          float e  = FAST_EXP2(u * u * kNegHL2e);  // v_exp_f32
          float pu = kPi * u;
          af[r][0] = __cosf(pu) * e;               // real feature (K_local even)
          af[r][1] = __sinf(pu) * e;               // imag feature (K_local odd)
        }

        // 8 independent accumulators -> no WMMA->WMMA RAW on the same regs
#pragma unroll
        for (int r = 0; r < kRowTiles; ++r)
#pragma unroll
          for (int t = 0; t < kColTiles; ++t)
            acc[r][t] = __builtin_amdgcn_wmma_f32_16x16x4_f32(
                false, af[r], false, bf[t], (short)0, acc[r][t], false, false);
      }
    }
  }

  // C/D layout: VGPR v -> M = v + 8*half, N = l16 (per 16x16 f32 tile)
#pragma unroll
  for (int r = 0; r < kRowTiles; ++r)
#pragma unroll
    for (int t = 0; t < kColTiles; ++t)
#pragma unroll
      for (int v = 0; v < 8; ++v) {
        int row = rowbase + r * 16 + v + 8 * half;
        out[(size_t)row * kOutDim + t * 16 + l16] = acc[r][t][v] + bv[t];
      }
}

// ---------------------------------------------------------------------------
extern "C" void kernel_launch(void* const* d_in, const int* in_sizes, int n_in,
                              void* d_out, int out_size, void* d_ws, size_t ws_size,
                              hipStream_t stream) {
  const float* x    = (const float*)d_in[0];   // (16384, 256) fp32
  const float* wc   = (const float*)d_in[1];   // (2, 64, 256, 16) fp32
  const float* bias = (const float*)d_in[2];   // (1, 64) fp32
  // d_in[3] = gridsize (static 16, baked into the kernels)
  float* out = (float*)d_out;                  // (16384, 64) fp32
  float* pw  = (float*)d_ws;                   // needs 2 MB of scratch

  pack_weights_kernel<<<kPackElems / 256, 256, 0, stream>>>(wc, pw);
  wavelet_kan_wmma_kernel<<<kB / kRowsPerWg, kWavesPerWg * 32, 0, stream>>>(
      x, pw, bias, out);
}